// PiecewiseRationalQuadraticCouplingTransform_16578573763425
// MI455X (gfx1250) — compile-verified
//
#include <hip/hip_runtime.h>

// ---------------------------------------------------------------------------
// Fused neural-spline coupling transform for gfx1250 (MI455X).
//   params = ResNet(ident) via V_WMMA_F32_16X16X4_F32 (full fp32)
//   y, lad = RQ-spline(trans, params) in VALU
// One wave owns 16 batch rows; 4 waves/block; h/t/params live in LDS (312 KB).
// Explicit addrspace(1)/(3) accessors -> global_load/ds_load (no flat), and
// double-buffered A/B fragments so weight loads overlap the WMMA pipe.
// ---------------------------------------------------------------------------

typedef __attribute__((ext_vector_type(2))) float v2f;
typedef __attribute__((ext_vector_type(8))) float v8f;

#define GLB __attribute__((address_space(1)))
#define SHM __attribute__((address_space(3)))
typedef SHM float lds_f;

#define B_TOTAL   131072
#define HID       256
#define NPAR      736          // 32 dims * 23 params
#define WAVE_LDS  19968        // floats/wave: 4096 (h) + 4096 (t) + 11776 (params)

// ---- address-space-pinned accessors ----
static __device__ inline v2f ldg_v2(const float* p) {
    return *(const GLB v2f*)(const GLB float*)p;
}
static __device__ inline float ldg_f(const float* p) {
    return *(const GLB float*)p;
}
static __device__ inline void stg_v2(float* p, v2f v) {
    *(GLB v2f*)(GLB float*)p = v;
}
static __device__ inline void stg_f(float* p, float v) {
    *(GLB float*)p = v;
}
static __device__ inline v2f ldl_v2(const lds_f* p) {
    return *(const SHM v2f*)p;
}

static __device__ inline v8f zero8() {
    v8f z;
#pragma unroll
    for (int e = 0; e < 8; ++e) z[e] = 0.0f;
    return z;
}

static __device__ inline v8f wmma4(v2f a, v2f b, v8f c) {
    return __builtin_amdgcn_wmma_f32_16x16x4_f32(
        false, a, false, b, (short)0, c, false, false);
}

// ---------------------------------------------------------------------------
// One GEMM pass: NT n-tiles (NT*16 cols), K = 4*KC reduction.
//   obuf[0:16, nt0*16 : nt0*16+NT*16] = relu?(sbuf) @ W^T + b [+ obuf]
// sbuf row stride = K, obuf row stride = OSTR, W is (N x K) row-major global.
// Double-buffered A (LDS) and B (global) fragments for load/WMMA overlap.
// ---------------------------------------------------------------------------
template <int NT, int KC, int OSTR>
static __device__ inline void gemmPass(const float* __restrict__ W,
                                       const float* __restrict__ bvec,
                                       const lds_f* __restrict__ sbuf,
                                       lds_f* __restrict__ obuf,
                                       int nt0, bool relu_in, bool res_add,
                                       int l16, int half) {
    constexpr int K = 4 * KC;
    v8f acc[NT];
#pragma unroll
    for (int i = 0; i < NT; ++i) acc[i] = zero8();

    const lds_f* arow  = sbuf + l16 * K + 2 * half;
    const float* wbase = W + (nt0 * 16 + l16) * K + 2 * half;

    v2f abuf[2];
    v2f bbuf[2][NT];
    abuf[0] = ldl_v2(arow);
#pragma unroll
    for (int nt = 0; nt < NT; ++nt) bbuf[0][nt] = ldg_v2(wbase + nt * 16 * K);

#pragma unroll 2
    for (int c = 0; c < KC; ++c) {
        const int cur = c & 1, nxt = cur ^ 1;
        if (c + 1 < KC) {                       // prefetch chunk c+1
            const int kn = 4 * (c + 1);
            abuf[nxt] = ldl_v2(arow + kn);
#pragma unroll
            for (int nt = 0; nt < NT; ++nt)
                bbuf[nxt][nt] = ldg_v2(wbase + nt * 16 * K + kn);
        }
        v2f a = abuf[cur];
        if (relu_in) { a.x = fmaxf(a.x, 0.0f); a.y = fmaxf(a.y, 0.0f); }
#pragma unroll
        for (int nt = 0; nt < NT; ++nt)
            acc[nt] = wmma4(a, bbuf[cur][nt], acc[nt]);
    }

#pragma unroll
    for (int nt = 0; nt < NT; ++nt) {
        const int n = (nt0 + nt) * 16 + l16;
        const float bias = ldg_f(bvec + n);
#pragma unroll
        for (int v = 0; v < 8; ++v) {
            const int row = v + 8 * half;
            lds_f* dst = obuf + row * OSTR + n;
            float val = acc[nt][v] + bias;
            if (res_add) val += *dst;
            *dst = val;
        }
    }
}

// ---- RQ spline for one scalar given its 23 params (uw[8], uh[8], ud[7]) ----
static __device__ inline void rq_spline(const lds_f* __restrict__ p, float xv,
                                        float& yout, float& ladout) {
    const float scale = 0.0625f;  // 1/sqrt(256)
    float uw[8], uh[8];
#pragma unroll
    for (int j = 0; j < 8; ++j) { uw[j] = p[j] * scale; uh[j] = p[8 + j] * scale; }

    float mw = uw[0], mh = uh[0];
#pragma unroll
    for (int j = 1; j < 8; ++j) { mw = fmaxf(mw, uw[j]); mh = fmaxf(mh, uh[j]); }
    float ew[8], eh[8], sw = 0.0f, sh = 0.0f;
#pragma unroll
    for (int j = 0; j < 8; ++j) {
        ew[j] = __expf(uw[j] - mw); sw += ew[j];
        eh[j] = __expf(uh[j] - mh); sh += eh[j];
    }
    const float rw = (1.0f - 0.008f) / sw;
    const float rh = (1.0f - 0.008f) / sh;
    float cw[9], ch[9];
    cw[0] = -3.0f; ch[0] = -3.0f;
    float aw = 0.0f, ah = 0.0f;
#pragma unroll
    for (int j = 0; j < 8; ++j) {
        aw += 0.001f + rw * ew[j]; cw[j + 1] = 6.0f * aw - 3.0f;
        ah += 0.001f + rh * eh[j]; ch[j + 1] = 6.0f * ah - 3.0f;
    }
    cw[8] = 3.0f; ch[8] = 3.0f;

    float dv[9];
    dv[0] = 1.0f; dv[8] = 1.0f;                // edges pinned at 1.0
#pragma unroll
    for (int j = 0; j < 7; ++j) {
        const float u = p[16 + j];
        const float sp = (u > 15.0f) ? u : __logf(1.0f + __expf(u));
        dv[j + 1] = 0.001f + sp;
    }

    const float xc = fminf(fmaxf(xv, -3.0f), 3.0f);

    float icw = cw[0], inw = cw[1] - cw[0];
    float ich = ch[0], inh = ch[1] - ch[0];
    float dlo = dv[0], dhi = dv[1];
#pragma unroll
    for (int j = 1; j < 8; ++j) {              // bin select -> cndmask chains
        const bool c = xc >= cw[j];
        icw = c ? cw[j] : icw;
        inw = c ? (cw[j + 1] - cw[j]) : inw;
        ich = c ? ch[j] : ich;
        inh = c ? (ch[j + 1] - ch[j]) : inh;
        dlo = c ? dv[j] : dlo;
        dhi = c ? dv[j + 1] : dhi;
    }

    const float delta = inh / inw;
    const float theta = (xc - icw) / inw;
    const float omt   = 1.0f - theta;
    const float tom   = theta * omt;
    const float num   = inh * (delta * theta * theta + dlo * tom);
    const float den   = delta + (dlo + dhi - 2.0f * delta) * tom;
    const float y     = ich + num / den;
    const float dnum  = delta * delta *
                        (dhi * theta * theta + 2.0f * delta * tom + dlo * omt * omt);
    const float lad   = __logf(dnum) - 2.0f * __logf(den);

    const bool inside = (xv >= -3.0f) && (xv <= 3.0f);
    yout   = inside ? y : xv;
    ladout = inside ? lad : 0.0f;
}

__global__ __launch_bounds__(128, 1)
void nsf_coupling_fused(const float* __restrict__ x,
                        const float* __restrict__ w_in, const float* __restrict__ b_in,
                        const float* __restrict__ w0a, const float* __restrict__ b0a,
                        const float* __restrict__ w1a, const float* __restrict__ b1a,
                        const float* __restrict__ w0b, const float* __restrict__ b0b,
                        const float* __restrict__ w1b, const float* __restrict__ b1b,
                        const float* __restrict__ w_out, const float* __restrict__ b_out,
                        float* __restrict__ out, float* __restrict__ lad_out) {
    extern __shared__ float dynlds[];
    const int wave = threadIdx.x >> 5;
    const int lane = threadIdx.x & 31;
    const int half = lane >> 4;
    const int l16  = lane & 15;

    lds_f* base = (lds_f*)(dynlds) + wave * WAVE_LDS;
    lds_f* hbuf = base;                    // 16 x 256
    lds_f* tbuf = base + 4096;             // 16 x 256 (also lad-reduction scratch)
    lds_f* pbuf = base + 8192;             // 16 x 736 (also ident staging at start)

    const int r0 = (blockIdx.x * 4 + wave) * 16;   // batch row base for this wave

    // ---- stage ident (odd cols of x) as 16x32 into pbuf region ----
    lds_f* identb = pbuf;
#pragma unroll
    for (int i = 0; i < 16; ++i) {
        v2f xin = ldg_v2(x + (size_t)(r0 + i) * 64 + 2 * lane);   // coalesced
        identb[i * 32 + lane] = xin.y;
    }
    __syncthreads();

    // ---- GEMM1: h = ident @ w_in^T + b_in  (K=32) ----
    gemmPass<8, 8, HID>(w_in, b_in, identb, hbuf, 0, false, false, l16, half);
    gemmPass<8, 8, HID>(w_in, b_in, identb, hbuf, 8, false, false, l16, half);
    __syncthreads();

    // ---- Residual block A ----
    gemmPass<8, 64, HID>(w0a, b0a, hbuf, tbuf, 0, true, false, l16, half);
    gemmPass<8, 64, HID>(w0a, b0a, hbuf, tbuf, 8, true, false, l16, half);
    __syncthreads();
    gemmPass<8, 64, HID>(w1a, b1a, tbuf, hbuf, 0, true, true, l16, half);
    gemmPass<8, 64, HID>(w1a, b1a, tbuf, hbuf, 8, true, true, l16, half);
    __syncthreads();

    // ---- Residual block B ----
    gemmPass<8, 64, HID>(w0b, b0b, hbuf, tbuf, 0, true, false, l16, half);
    gemmPass<8, 64, HID>(w0b, b0b, hbuf, tbuf, 8, true, false, l16, half);
    __syncthreads();
    gemmPass<8, 64, HID>(w1b, b1b, tbuf, hbuf, 0, true, true, l16, half);
    gemmPass<8, 64, HID>(w1b, b1b, tbuf, hbuf, 8, true, true, l16, half);
    __syncthreads();

    // ---- Output GEMM: params = h @ w_out^T + b_out  (N = 736 = 5*8+6 tiles) ----
    gemmPass<8, 64, NPAR>(w_out, b_out, hbuf, pbuf,  0, false, false, l16, half);
    gemmPass<8, 64, NPAR>(w_out, b_out, hbuf, pbuf,  8, false, false, l16, half);
    gemmPass<8, 64, NPAR>(w_out, b_out, hbuf, pbuf, 16, false, false, l16, half);
    gemmPass<8, 64, NPAR>(w_out, b_out, hbuf, pbuf, 24, false, false, l16, half);
    gemmPass<8, 64, NPAR>(w_out, b_out, hbuf, pbuf, 32, false, false, l16, half);
    gemmPass<6, 64, NPAR>(w_out, b_out, hbuf, pbuf, 40, false, false, l16, half);
    __syncthreads();

    // ---- Spline: lane = transform-dim d, loop over this wave's 16 rows ----
#pragma unroll 1
    for (int i = 0; i < 16; ++i) {
        const size_t rowoff = (size_t)(r0 + i) * 64 + 2 * lane;
        const v2f xin = ldg_v2(x + rowoff);            // (trans, ident) coalesced
        float y, lad;
        rq_spline(pbuf + i * NPAR + lane * 23, xin.x, y, lad);
        v2f o; o.x = y; o.y = xin.y;                   // interleave y with identity
        stg_v2(out + rowoff, o);
        tbuf[i * 32 + lane] = lad;                     // stage for reduction
    }
    __syncthreads();

    // ---- log-det reduction: per-row sum over 32 dims ----
    if (lane < 16) {
        const lds_f* r = tbuf + lane * 32;
        float s = 0.0f;
#pragma unroll
        for (int j = 0; j < 32; ++j) s += r[j];
        stg_f(lad_out + r0 + lane, s);
    }
}

extern "C" void kernel_launch(void* const* d_in, const int* in_sizes, int n_in,
                              void* d_out, int out_size, void* d_ws, size_t ws_size,
                              hipStream_t stream) {
    (void)in_sizes; (void)n_in; (void)out_size; (void)d_ws; (void)ws_size;
    const float* x     = (const float*)d_in[0];
    const float* w_in  = (const float*)d_in[1];
    const float* b_in  = (const float*)d_in[2];
    const float* w0a   = (const float*)d_in[3];
    const float* b0a   = (const float*)d_in[4];
    const float* w1a   = (const float*)d_in[5];
    const float* b1a   = (const float*)d_in[6];
    const float* w0b   = (const float*)d_in[7];
    const float* b0b   = (const float*)d_in[8];
    const float* w1b   = (const float*)d_in[9];
    const float* b1b   = (const float*)d_in[10];
    const float* w_out = (const float*)d_in[11];
    const float* b_out = (const float*)d_in[12];

    float* out = (float*)d_out;
    float* lad = out + (size_t)B_TOTAL * 64;

    const int blocks = B_TOTAL / 64;                             // 64 rows/block
    const size_t shmem = (size_t)4 * WAVE_LDS * sizeof(float);   // 312 KB < 320 KB
    nsf_coupling_fused<<<blocks, 128, shmem, stream>>>(
        x, w_in, b_in, w0a, b0a, w1a, b1a, w0b, b0b, w1b, b1b, w_out, b_out,
        out, lad);
}